// AttentionContext_14147622273492
// MI455X (gfx1250) — compile-verified
//
#include <hip/hip_runtime.h>
#include <hip/hip_bf16.h>
#include <stddef.h>

// Problem dims (fixed by the reference)
#define BB 64
#define TT 2048
#define HH 512
#define SS 512
#define KK 128
#define VV 128
#define EPS 1e-12f
#define NWB 64            // online-softmax waves per batch row
#define TW (TT / NWB)     // 32 t-rows per wave

typedef float v2f __attribute__((ext_vector_type(2)));
typedef float v8f __attribute__((ext_vector_type(8)));

// ---------------------------------------------------------------------------
// WMMA f32 16x16x4 tile GEMMs. One wave (32 threads) per 16x16 output tile.
// A layout (16x4 f32): lane l holds M=l&15; VGPR v holds K = 2*(l>>4) + v.
// B layout (4x16 f32): lane l holds N=l&15; VGPR v holds K = 2*(l>>4) + v.
// C/D layout: lane l holds N=l&15; VGPR r holds M = r + 8*(l>>4).
// ---------------------------------------------------------------------------

// query[m,n] = sum_s d[m,s]*Ws[n,s] + bs[n]   (64x128, K=512)
__global__ __launch_bounds__(32) void k_query_wmma(
    const float* __restrict__ d, const float* __restrict__ Ws,
    const float* __restrict__ bs, float* __restrict__ query) {
  const int tn = blockIdx.x, tm = blockIdx.y;
  const int l = threadIdx.x, half = l >> 4, l16 = l & 15;
  const int m = tm * 16 + l16, n = tn * 16 + l16;
  v8f c = {};
  for (int k = 0; k < SS; k += 4) {
    const int kb = k + half * 2;
    v2f a, b;
    a.x = d[m * SS + kb];      a.y = d[m * SS + kb + 1];
    b.x = Ws[n * SS + kb];     b.y = Ws[n * SS + kb + 1];   // B[k][n] = Ws[n][k]
    c = __builtin_amdgcn_wmma_f32_16x16x4_f32(false, a, false, b, (short)0, c,
                                              false, false);
  }
  const float bias = bs[n];
#pragma unroll
  for (int r = 0; r < 8; ++r) {
    const int mr = tm * 16 + r + half * 8;
    query[mr * KK + n] = c[r] + bias;
  }
}

// qh[m,n] = sum_k query[m,k]*Wh[k,n]          (64x512, K=128)
__global__ __launch_bounds__(32) void k_qh_wmma(
    const float* __restrict__ query, const float* __restrict__ Wh,
    float* __restrict__ qh) {
  const int tn = blockIdx.x, tm = blockIdx.y;
  const int l = threadIdx.x, half = l >> 4, l16 = l & 15;
  const int m = tm * 16 + l16, n = tn * 16 + l16;
  v8f c = {};
  for (int k = 0; k < KK; k += 4) {
    const int kb = k + half * 2;
    v2f a, b;
    a.x = query[m * KK + kb];       a.y = query[m * KK + kb + 1];
    b.x = Wh[kb * HH + n];          b.y = Wh[(kb + 1) * HH + n]; // B[k][n] = Wh[k][n]
    c = __builtin_amdgcn_wmma_f32_16x16x4_f32(false, a, false, b, (short)0, c,
                                              false, false);
  }
#pragma unroll
  for (int r = 0; r < 8; ++r) {
    const int mr = tm * 16 + r + half * 8;
    qh[mr * HH + n] = c[r];
  }
}

// ctx[m,n] = sum_h ch[m,h]*Wv[n,h] + bv[n]*satt[m]   (64x128, K=512)
__global__ __launch_bounds__(32) void k_context_wmma(
    const float* __restrict__ ch, const float* __restrict__ Wv,
    const float* __restrict__ bv, const float* __restrict__ satt,
    float* __restrict__ ctx) {
  const int tn = blockIdx.x, tm = blockIdx.y;
  const int l = threadIdx.x, half = l >> 4, l16 = l & 15;
  const int m = tm * 16 + l16, n = tn * 16 + l16;
  v8f c = {};
  for (int k = 0; k < HH; k += 4) {
    const int kb = k + half * 2;
    v2f a, b;
    a.x = ch[m * HH + kb];     a.y = ch[m * HH + kb + 1];
    b.x = Wv[n * HH + kb];     b.y = Wv[n * HH + kb + 1];   // B[k][n] = Wv[n][k]
    c = __builtin_amdgcn_wmma_f32_16x16x4_f32(false, a, false, b, (short)0, c,
                                              false, false);
  }
  const float bvn = bv[n];
#pragma unroll
  for (int r = 0; r < 8; ++r) {
    const int mr = tm * 16 + r + half * 8;
    ctx[mr * VV + n] = c[r] + bvn * satt[mr];
  }
}

// qb[b] = query[b,:] . bh
__global__ __launch_bounds__(64) void k_qb(const float* __restrict__ query,
                                           const float* __restrict__ bh,
                                           float* __restrict__ qb) {
  const int b = threadIdx.x;
  float s = 0.f;
  for (int k = 0; k < KK; ++k) s += query[b * KK + k] * bh[k];
  qb[b] = s;
}

// ---------------------------------------------------------------------------
// SINGLE streaming pass over listener_output (flash-style online softmax):
// one wave handles TW=32 consecutive t-rows of one batch. Per row:
//   e = L[b,t,:].qh[b,:] + qb[b]       (float4 loads + butterfly shfl reduce)
//   spill e to scratch (needed for the attention output pass)
//   online update of m, Z_all, Z_valid and c_h += exp(e-m)*L[t,:]
// Row data is reused from registers for the c_h accumulation -> listener is
// read exactly ONCE (256 MB ~ 11us at 23.3 TB/s), the true roofline.
// ---------------------------------------------------------------------------
__global__ __launch_bounds__(256) void k_fused(
    const float* __restrict__ L, const float* __restrict__ qh,
    const float* __restrict__ qb, const int* __restrict__ lens,
    float* __restrict__ energy, float* __restrict__ wm,
    float* __restrict__ wza, float* __restrict__ wzv,
    float* __restrict__ chpart) {
  const int b = blockIdx.y;
  const int wave = (int)(threadIdx.x >> 5), lane = (int)(threadIdx.x & 31);
  const int widx = blockIdx.x * 8 + wave;       // 0..NWB-1 within batch b
  const int t0 = widx * TW;
  const int len = lens[b];
  const float qbb = qb[b];

  const float4* Q = (const float4*)(qh + b * HH);
  const float4 q0 = Q[lane], q1 = Q[32 + lane], q2 = Q[64 + lane],
               q3 = Q[96 + lane];
  float4 c0 = {0.f, 0.f, 0.f, 0.f}, c1 = c0, c2 = c0, c3 = c0;
  float m_run = -1e30f, za = 0.f, zv = 0.f;

  const float4* Lb = (const float4*)(L + ((size_t)b * TT + t0) * HH);
  for (int j = 0; j < TW; ++j) {
    const float4* Lr = Lb + (size_t)j * (HH / 4);
    const float4 x0 = Lr[lane], x1 = Lr[32 + lane], x2 = Lr[64 + lane],
                 x3 = Lr[96 + lane];
    float acc = x0.x * q0.x + x0.y * q0.y + x0.z * q0.z + x0.w * q0.w;
    acc += x1.x * q1.x + x1.y * q1.y + x1.z * q1.z + x1.w * q1.w;
    acc += x2.x * q2.x + x2.y * q2.y + x2.z * q2.z + x2.w * q2.w;
    acc += x3.x * q3.x + x3.y * q3.y + x3.z * q3.z + x3.w * q3.w;
#pragma unroll
    for (int off = 16; off; off >>= 1) acc += __shfl_xor(acc, off, 32);
    const float e = acc + qbb;                 // wave-uniform
    if (lane == 0) energy[b * TT + t0 + j] = e;
    if (e > m_run) {                           // wave-uniform branch
      const float s = __expf(m_run - e);       // first iter: underflows to 0
      za *= s; zv *= s;
      c0.x *= s; c0.y *= s; c0.z *= s; c0.w *= s;
      c1.x *= s; c1.y *= s; c1.z *= s; c1.w *= s;
      c2.x *= s; c2.y *= s; c2.z *= s; c2.w *= s;
      c3.x *= s; c3.y *= s; c3.z *= s; c3.w *= s;
      m_run = e;
    }
    const float w = __expf(e - m_run);
    za += w;                                   // Z_all includes masked rows
    const bool valid = (b == 0) || ((t0 + j) < len);
    const float wv = valid ? w : 0.f;
    zv += wv;
    c0.x += wv * x0.x; c0.y += wv * x0.y; c0.z += wv * x0.z; c0.w += wv * x0.w;
    c1.x += wv * x1.x; c1.y += wv * x1.y; c1.z += wv * x1.z; c1.w += wv * x1.w;
    c2.x += wv * x2.x; c2.y += wv * x2.y; c2.z += wv * x2.z; c2.w += wv * x2.w;
    c3.x += wv * x3.x; c3.y += wv * x3.y; c3.z += wv * x3.z; c3.w += wv * x3.w;
  }
  const int g = b * NWB + widx;
  if (lane == 0) { wm[g] = m_run; wza[g] = za; wzv[g] = zv; }
  float4* P = (float4*)(chpart + (size_t)g * HH);
  P[lane] = c0; P[32 + lane] = c1; P[64 + lane] = c2; P[96 + lane] = c3;
}

// ---------------------------------------------------------------------------
// Per-batch combine of the NWB wave partials (fixed order -> deterministic):
//   m = max m_w;  Z = sum Z_w*exp(m_w-m);  c_h = sum exp(m_w-m)*c_h_w
//   att[b,t] = mask*exp(e-m)/max(Zv, EPS*Za)  (reference softmax->mask->L1)
// ---------------------------------------------------------------------------
__global__ __launch_bounds__(128) void k_combine(
    const float* __restrict__ wm, const float* __restrict__ wza,
    const float* __restrict__ wzv, const float* __restrict__ chpart,
    float* __restrict__ ch, float* __restrict__ m_arr,
    float* __restrict__ invd, float* __restrict__ satt) {
  const int b = blockIdx.x, tid = threadIdx.x;
  __shared__ float sco[NWB];
  __shared__ float sbrd[2];
  if (tid == 0) {
    float m = -1e30f;
    for (int w = 0; w < NWB; ++w) m = fmaxf(m, wm[b * NWB + w]);
    sbrd[0] = m;
  }
  __syncthreads();
  const float m = sbrd[0];
  if (tid < NWB) sco[tid] = __expf(wm[b * NWB + tid] - m);
  __syncthreads();
  if (tid == 0) {
    float za = 0.f, zv = 0.f;
    for (int w = 0; w < NWB; ++w) {
      za += wza[b * NWB + w] * sco[w];
      zv += wzv[b * NWB + w] * sco[w];
    }
    const float inv = 1.0f / fmaxf(zv, EPS * za);
    m_arr[b] = m; invd[b] = inv; satt[b] = zv * inv;
    sbrd[1] = inv;
  }
  __syncthreads();
  const float inv = sbrd[1];
  float4 acc = {0.f, 0.f, 0.f, 0.f};
  for (int w = 0; w < NWB; ++w) {
    const float4 p = ((const float4*)(chpart + (size_t)(b * NWB + w) * HH))[tid];
    const float cw = sco[w];
    acc.x += cw * p.x; acc.y += cw * p.y; acc.z += cw * p.z; acc.w += cw * p.w;
  }
  float4 r; r.x = acc.x * inv; r.y = acc.y * inv; r.z = acc.z * inv;
  r.w = acc.w * inv;
  ((float4*)(ch + b * HH))[tid] = r;
}

// Final attention values into d_out (second output, shape B x 1 x T)
__global__ __launch_bounds__(256) void k_att(
    const float* __restrict__ energy, const int* __restrict__ lens,
    const float* __restrict__ m_arr, const float* __restrict__ invd,
    float* __restrict__ att) {
  const int idx = blockIdx.x * 256 + threadIdx.x;
  const int b = idx >> 11, t = idx & (TT - 1);
  const bool valid = (b == 0) || (t < lens[b]);
  att[idx] = valid ? __expf(energy[idx] - m_arr[b]) * invd[b] : 0.0f;
}

// ---------------------------------------------------------------------------
extern "C" void kernel_launch(void* const* d_in, const int* in_sizes, int n_in,
                              void* d_out, int out_size, void* d_ws,
                              size_t ws_size, hipStream_t stream) {
  (void)in_sizes; (void)n_in; (void)out_size; (void)ws_size;
  const float* d_state = (const float*)d_in[0];   // (B,S)
  const float* L       = (const float*)d_in[1];   // (B,T,H)
  const int*   lens    = (const int*)d_in[2];     // (B,)
  const float* Ws      = (const float*)d_in[3];   // (K,S)
  const float* bs      = (const float*)d_in[4];   // (K,)
  const float* Wh      = (const float*)d_in[5];   // (K,H)
  const float* bh      = (const float*)d_in[6];   // (K,)
  const float* Wv      = (const float*)d_in[7];   // (V,H)
  const float* bv      = (const float*)d_in[8];   // (V,)

  float* out_ctx = (float*)d_out;                 // (B,V)
  float* out_att = out_ctx + BB * VV;             // (B,1,T)

  float* ws = (float*)d_ws;
  float* query   = ws;                      // B*K        = 8192
  float* qh      = query + BB * KK;         // B*H        = 32768
  float* qb      = qh + BB * HH;            // B          = 64
  float* energy  = qb + BB;                 // B*T        = 131072
  float* wm      = energy + BB * TT;        // B*NWB      = 4096
  float* wza     = wm + BB * NWB;           // 4096
  float* wzv     = wza + BB * NWB;          // 4096
  float* chpart  = wzv + BB * NWB;          // B*NWB*H    = 2097152 (8 MB)
  float* ch      = chpart + (size_t)BB * NWB * HH; // B*H = 32768
  float* m_arr   = ch + BB * HH;            // B
  float* invd    = m_arr + BB;              // B
  float* satt    = invd + BB;               // B   (~9.2 MB scratch total)

  // 1) query = d_state @ Ws^T + bs          [WMMA f32 16x16x4]
  k_query_wmma<<<dim3(KK / 16, BB / 16), 32, 0, stream>>>(d_state, Ws, bs, query);
  // 2) qh = query @ Wh                      [WMMA f32 16x16x4]
  k_qh_wmma<<<dim3(HH / 16, BB / 16), 32, 0, stream>>>(query, Wh, qh);
  //    qb = query . bh
  k_qb<<<1, BB, 0, stream>>>(query, bh, qb);
  // 3) fused single pass: energy + online softmax + weighted c_h partials
  k_fused<<<dim3(NWB / 8, BB), 256, 0, stream>>>(L, qh, qb, lens, energy, wm,
                                                 wza, wzv, chpart);
  // 4) per-batch combine (deterministic fixed-order reduction)
  k_combine<<<BB, 128, 0, stream>>>(wm, wza, wzv, chpart, ch, m_arr, invd, satt);
  // 5) attention output
  k_att<<<(BB * TT) / 256, 256, 0, stream>>>(energy, lens, m_arr, invd, out_att);
  // 6) context = c_h @ Wv^T + bv * sum(att) [WMMA f32 16x16x4]
  k_context_wmma<<<dim3(VV / 16, BB / 16), 32, 0, stream>>>(ch, Wv, bv, satt, out_ctx);
}